// MLPKANlayer_71794673320508
// MI455X (gfx1250) — compile-verified
//
#include <hip/hip_runtime.h>

// MI455X / gfx1250, wave32. fp32 WMMA (V_WMMA_F32_16X16X4_F32), reference
// numerics. Per wave: one output column `o`, 64 batch rows (four 16-wide
// N tiles), loop over 128 input features i (subnet n = i*128 + o).
//
//   layer2:  h2^T = W2' (A, M rows 0-7 = h0-7, rows 8,9,12,13 = h2,3,6,7)
//                   x h1^T (B, N=batch) + b2' (C, same row remap)
//   layer3:  acc  += W3 (A, broadcast to all rows) x h2^T (B)
//
// W2-row replication makes layer2's D land *exactly* in layer3's B-operand
// layout (vgprs 0,1,4,5: lower lanes = h0,1,4,5, upper lanes = h2,3,6,7):
// zero cross-lane ops. acc's C operand carries the sum over all 128 subnets.
// ReLU on WMMA outputs is raw inline asm (single v_max_num_f32) so the
// backend cannot insert a canonicalize (max x,x) before it. The C operand's
// dead elements (2,3,6,7) are undef via shufflevector so no movs are spent
// materializing them.

typedef float v2f __attribute__((ext_vector_type(2)));
typedef float v4f __attribute__((ext_vector_type(4)));
typedef float v8f __attribute__((ext_vector_type(8)));

#define IN_SZ   128
#define OUT_SZ  128
#define B_SZ    1024
#define NT      4                // 16-wide batch tiles per wave
#define ROWS    (16 * NT)        // 64 batch rows per wave
#define XPITCH  132              // LDS row pitch: 16B-aligned, conflict-free

static __device__ inline v8f wmma4(v2f a, v2f b, v8f c) {
  // (neg_a, A, neg_b, B, c_mod, C, reuse_a, reuse_b)
  return __builtin_amdgcn_wmma_f32_16x16x4_f32(false, a, false, b, (short)0, c,
                                               false, false);
}

// single-instruction ReLU the compiler can't "canonicalize"
static __device__ inline float relu1(float x) {
  float r;
  asm("v_max_num_f32 %0, %1, 0" : "=v"(r) : "v"(x));
  return r;
}

__global__ __launch_bounds__(256) void mlpkan_kernel(
    const float* __restrict__ x,  const float* __restrict__ W1,
    const float* __restrict__ b1, const float* __restrict__ W2,
    const float* __restrict__ b2, const float* __restrict__ W3,
    const float* __restrict__ b3, float* __restrict__ out) {
  __shared__ float xs[ROWS * XPITCH];

  const int tid = threadIdx.x;
  const int b0  = (blockIdx.x >> 4) * ROWS;  // 8 waves/block share this slab

  for (int t = tid; t < ROWS * IN_SZ; t += 256) {
    const int r = t >> 7, c = t & (IN_SZ - 1);
    xs[r * XPITCH + c] = x[(b0 + r) * IN_SZ + c];
  }
  __syncthreads();

  const int lane = tid & 31;
  const int wave = blockIdx.x * 8 + (tid >> 5);
  const int o    = wave & (OUT_SZ - 1);  // output column for this wave
  const int lm   = lane & 15;
  const int hi   = lane >> 4;            // upper half-wave?
  const int kb   = hi * 2;               // K base per A/B layout

  // A2 row map: m<8 -> h=m ; m=8,9,12,13 -> h=m-6 (duplicated h2,3,6,7);
  // m=10,11,14,15 unused (clamped in-bounds).
  int hrow = (lm < 8) ? lm : (lm - 6);
  if (hrow > 7) hrow = 7;
  const int h8 = hrow * 8;

  v8f acc[NT];
#pragma unroll
  for (int t = 0; t < NT; ++t)
    acc[t] = (v8f){0.f, 0.f, 0.f, 0.f, 0.f, 0.f, 0.f, 0.f};
  float sbias = 0.f;

  for (int i0 = 0; i0 < IN_SZ; i0 += 4) {
    // one b128 LDS read per tile covers x for 4 consecutive i's
    v4f sx[NT];
#pragma unroll
    for (int t = 0; t < NT; ++t)
      sx[t] = *(const v4f*)(xs + (lm + 16 * t) * XPITCH + i0);

#pragma unroll
    for (int ii = 0; ii < 4; ++ii) {
      const int n  = (i0 + ii) * OUT_SZ + o;
      const int n8 = n * 8;

      const v2f w1a = *(const v2f*)(W1 + n8 + kb);
      const v2f w1b = *(const v2f*)(W1 + n8 + kb + 4);
      const v2f c1a = *(const v2f*)(b1 + n8 + kb);
      const v2f c1b = *(const v2f*)(b1 + n8 + kb + 4);
      const v2f w2a = *(const v2f*)(W2 + n * 64 + h8 + kb);      // A2, K=0..3
      const v2f w2b = *(const v2f*)(W2 + n * 64 + h8 + kb + 4);  // A2, K=4..7
      const v2f w3a = *(const v2f*)(W3 + n8 + kb);               // A3
      const v2f w3b = *(const v2f*)(W3 + n8 + kb + 4);
      // C2 bias with the same row remap: lower lanes b2[v], upper b2[v+2].
      // Elements 2,3,6,7 feed dead D elements -> leave them undef.
      const v2f cba = *(const v2f*)(b2 + n8 + kb);
      const v2f cbb = *(const v2f*)(b2 + n8 + kb + 4);
      const v8f cb  = __builtin_shufflevector(cba, cbb,
                                              0, 1, -1, -1, 2, 3, -1, -1);
      sbias += b3[n];

      __builtin_prefetch(W2 + (n + OUT_SZ) * 64, 0, 3);  // next subnet's W2

#pragma unroll
      for (int t = 0; t < NT; ++t) {
        const float sv = sx[t][ii];

        // layer 1: this lane's h1 values for B2 slots (i2 = kb+{0,1,4,5})
        v2f ha, hb;
        ha.x = fmaxf(fmaf(sv, w1a.x, c1a.x), 0.f);
        ha.y = fmaxf(fmaf(sv, w1a.y, c1a.y), 0.f);
        hb.x = fmaxf(fmaf(sv, w1b.x, c1b.x), 0.f);
        hb.y = fmaxf(fmaf(sv, w1b.y, c1b.y), 0.f);

        // layer 2 (two K=4 steps); only d[0,1,4,5] are live afterwards
        v8f d = wmma4(w2a, ha, cb);
        d     = wmma4(w2b, hb, d);
        v2f g0, g1;
        g0.x = relu1(d[0]);
        g0.y = relu1(d[1]);
        g1.x = relu1(d[4]);
        g1.y = relu1(d[5]);

        // layer 3: accumulate subnet output into acc (sum over i for free)
        acc[t] = wmma4(w3a, g0, acc[t]);
        acc[t] = wmma4(w3b, g1, acc[t]);
      }
    }
  }

  // every row of acc equals o_sum[b = lane%16]; halves are duplicates
  if (hi == 0) {
#pragma unroll
    for (int t = 0; t < NT; ++t)
      out[(b0 + 16 * t + lm) * OUT_SZ + o] = acc[t][0] + sbias;
  }
}

extern "C" void kernel_launch(void* const* d_in, const int* in_sizes, int n_in,
                              void* d_out, int out_size, void* d_ws, size_t ws_size,
                              hipStream_t stream) {
  const float* x  = (const float*)d_in[0];
  const float* W1 = (const float*)d_in[1];
  const float* b1 = (const float*)d_in[2];
  const float* W2 = (const float*)d_in[3];
  const float* b2 = (const float*)d_in[4];
  const float* W3 = (const float*)d_in[5];
  const float* b3 = (const float*)d_in[6];
  float* out = (float*)d_out;

  // waves = 128 (o) * 16 (batch/64) = 2048; 8 waves per 256-thread block
  dim3 grid((OUT_SZ * (B_SZ / ROWS)) / 8), block(256);
  mlpkan_kernel<<<grid, block, 0, stream>>>(x, W1, b1, W2, b2, W3, b3, out);
}